// AttentionPair_49538152792199
// MI455X (gfx1250) — compile-verified
//
#include <hip/hip_runtime.h>
#include <math.h>

typedef float v2f __attribute__((ext_vector_type(2)));
typedef float v8f __attribute__((ext_vector_type(8)));

#define B_   64
#define S_   512
#define DV_  1024
#define DA_  512
#define DM_  1024
#define LDSA_STRIDE 260                 // 256 + 4 pad -> conflict-free b64 LDS reads
#define ABUF_FLOATS (32 * LDSA_STRIDE)  // one double-buffer slot: 32 rows x 256 K (+pad)

// ---------------- Kernel 1: t1 = vector @ W_vec  [B, DA] ----------------
__global__ __launch_bounds__(256) void k_t1(const float* __restrict__ vec,
                                            const float* __restrict__ Wv,
                                            float* __restrict__ t1) {
    const int b = blockIdx.y;
    const int a = blockIdx.x * 256 + threadIdx.x;
    const float* v = vec + (size_t)b * DV_;
    float acc = 0.f;
#pragma unroll 4
    for (int k = 0; k < DV_; ++k)
        acc += v[k] * Wv[(size_t)k * DA_ + a];
    t1[b * DA_ + a] = acc;
}

// CDNA5 async global->LDS copy (tracked by ASYNCcnt), 16B per lane.
__device__ __forceinline__ void async_b128(unsigned lds_byte, const float* g) {
    asm volatile("global_load_async_to_lds_b128 %0, %1, off"
                 :: "v"(lds_byte), "v"(g) : "memory");
}

// ------- Kernel 2: fused GEMM + relu + dot(w_attn) -> logits[B*S] -------
// One block = 32 rows of (b,s). 8 waves x 4 N-tiles cover DA=512; each wave
// keeps two M-subtile accumulators so every B fragment feeds 2 WMMAs.
// K=1024 looped as 4 chunks of 256, double-buffered in LDS via async copies.
__global__ __launch_bounds__(256) void k_logits(const float* __restrict__ matrix,
                                                const float* __restrict__ Wm,
                                                const float* __restrict__ t1,
                                                const float* __restrict__ w_attn,
                                                float* __restrict__ logits) {
    extern __shared__ float smem[];
    float* bufs      = smem;                 // 2 * ABUF_FLOATS
    float* lds_logit = smem + 2 * ABUF_FLOATS;  // 32 floats

    const int tid  = threadIdx.x;
    const int lane = tid & 31;
    const int wv   = tid >> 5;       // wave 0..7
    const int half = lane >> 4;      // 0: lanes 0-15, 1: lanes 16-31
    const int ln   = lane & 15;

    const int m0 = blockIdx.x * 32;  // first global row (b*S + s)
    const int b  = m0 >> 9;          // 512 rows per batch; 32 | 512 so single b

    if (tid < 32) lds_logit[tid] = 0.f;

    v8f acc0[4] = {};                // rows m0 + 0..15
    v8f acc1[4] = {};                // rows m0 + 16..31

    // per-thread staging geometry: 32 rows x 256 floats, 32 floats/thread
    const int srow = tid >> 3;
    const int sc0  = (tid & 7) * 32;
    const size_t rowBase = (size_t)m0 * DM_;
    const float* gsrc_row = matrix + rowBase + (size_t)srow * DM_ + sc0;

    // issue chunk 0 into buffer 0
    {
        const float* g = gsrc_row;                       // kc = 0
        unsigned l = (unsigned)(uintptr_t)&bufs[srow * LDSA_STRIDE + sc0];
#pragma unroll
        for (int j = 0; j < 8; ++j) async_b128(l + j * 16, g + j * 4);
    }

    for (int kc = 0; kc < 4; ++kc) {
        if (kc < 3) {   // prefetch next chunk into the other buffer
            const float* g = gsrc_row + (kc + 1) * 256;
            unsigned l = (unsigned)(uintptr_t)
                &bufs[((kc + 1) & 1) * ABUF_FLOATS + srow * LDSA_STRIDE + sc0];
#pragma unroll
            for (int j = 0; j < 8; ++j) async_b128(l + j * 16, g + j * 4);
            asm volatile("s_wait_asynccnt 0x8" ::: "memory");  // chunk kc landed
        } else {
            asm volatile("s_wait_asynccnt 0x0" ::: "memory");
        }
        __syncthreads();

        const float* abuf  = bufs + (kc & 1) * ABUF_FLOATS;
        const float* wbase = Wm + (size_t)(kc * 256) * DA_;
#pragma unroll 2
        for (int st = 0; st < 64; ++st) {
            const int kk = st * 4 + 2 * half;           // K-pair per lane half
            // A fragments (ISA 16x4 layout): lane ln holds row M, K = kk,kk+1
            v2f a0 = *(const v2f*)&abuf[ln * LDSA_STRIDE + kk];
            v2f a1 = *(const v2f*)&abuf[(ln + 16) * LDSA_STRIDE + kk];
            const float* bp = wbase + (size_t)kk * DA_;
#pragma unroll
            for (int nt = 0; nt < 4; ++nt) {
                const int n = (wv * 4 + nt) * 16 + ln;
                v2f bf;
                bf.x = bp[n];          // W_mat[k    ][n]
                bf.y = bp[DA_ + n];    // W_mat[k + 1][n]
                acc0[nt] = __builtin_amdgcn_wmma_f32_16x16x4_f32(
                    false, a0, false, bf, (short)0, acc0[nt], false, false);
                acc1[nt] = __builtin_amdgcn_wmma_f32_16x16x4_f32(
                    false, a1, false, bf, (short)0, acc1[nt], false, false);
            }
        }
        __syncthreads();   // buffer (kc&1) free for reuse next iteration
    }

    // Epilogue: logits[M] = sum_n relu(C[M][n] + t1[b][n]) * w_attn[n]
    const float* t1b = t1 + b * DA_;
#pragma unroll
    for (int nt = 0; nt < 4; ++nt) {
        const int n  = (wv * 4 + nt) * 16 + ln;
        const float t1v = t1b[n];
        const float wa  = w_attn[n];
        float s0[8], s1[8];
#pragma unroll
        for (int r = 0; r < 8; ++r) {
            float x0 = acc0[nt][r] + t1v;   // C layout: VGPR r -> M = r + 8*half
            float x1 = acc1[nt][r] + t1v;
            s0[r] = (x0 > 0.f ? x0 : 0.f) * wa;
            s1[r] = (x1 > 0.f ? x1 : 0.f) * wa;
        }
        // reduce over the 16 lanes holding N=0..15 of this tile
#pragma unroll
        for (int r = 0; r < 8; ++r) {
#pragma unroll
            for (int off = 8; off > 0; off >>= 1) {
                s0[r] += __shfl_xor(s0[r], off, 16);
                s1[r] += __shfl_xor(s1[r], off, 16);
            }
        }
        if (ln == 0) {
#pragma unroll
            for (int r = 0; r < 8; ++r) {
                atomicAdd(&lds_logit[r + 8 * half],      s0[r]);
                atomicAdd(&lds_logit[16 + r + 8 * half], s1[r]);
            }
        }
    }
    __syncthreads();
    if (tid < 32) logits[m0 + tid] = lds_logit[tid];
}

// ---------------- Kernel 3: global max of logits ----------------
__global__ __launch_bounds__(1024) void k_max(const float* __restrict__ logits,
                                              float* __restrict__ maxout) {
    __shared__ float red[1024];
    const int t = threadIdx.x;
    float m = -INFINITY;
    for (int i = t; i < B_ * S_; i += 1024) m = fmaxf(m, logits[i]);
    red[t] = m;
    __syncthreads();
    for (int off = 512; off > 0; off >>= 1) {
        if (t < off) red[t] = fmaxf(red[t], red[t + off]);
        __syncthreads();
    }
    if (t == 0) maxout[0] = red[0];
}

// --------- Kernel 4: masked softmax rows -> attentions [B, S] ---------
__global__ __launch_bounds__(512) void k_soft(const float* __restrict__ logits,
                                              const float* __restrict__ maxv,
                                              const int* __restrict__ lens,
                                              float* __restrict__ att_out) {
    __shared__ float red[512];
    const int b = blockIdx.x, s = threadIdx.x;
    float e = expf(logits[b * S_ + s] - maxv[0]);
    if (s >= lens[b]) e = 0.f;
    red[s] = e;
    __syncthreads();
    for (int off = 256; off > 0; off >>= 1) {
        if (s < off) red[s] += red[s + off];
        __syncthreads();
    }
    att_out[b * S_ + s] = e / red[0];
}

// --- Kernel 5: representations[b,d] = sum_s matrix[b,s,d]*att[b,s] ---
__global__ __launch_bounds__(256) void k_repr(const float* __restrict__ matrix,
                                              const float* __restrict__ att,
                                              float* __restrict__ rep) {
    __shared__ float a_s[S_];
    const int b  = blockIdx.x >> 2;
    const int d  = (blockIdx.x & 3) * 256 + threadIdx.x;
    for (int i = threadIdx.x; i < S_; i += 256) a_s[i] = att[b * S_ + i];
    __syncthreads();
    const float* mb = matrix + (size_t)b * S_ * DM_ + d;
    float a0 = 0.f, a1 = 0.f, a2 = 0.f, a3 = 0.f;
    for (int s = 0; s < S_; s += 4) {
        a0 += mb[(size_t)(s + 0) * DM_] * a_s[s + 0];
        a1 += mb[(size_t)(s + 1) * DM_] * a_s[s + 1];
        a2 += mb[(size_t)(s + 2) * DM_] * a_s[s + 2];
        a3 += mb[(size_t)(s + 3) * DM_] * a_s[s + 3];
    }
    rep[b * DM_ + d] = (a0 + a1) + (a2 + a3);
}

extern "C" void kernel_launch(void* const* d_in, const int* in_sizes, int n_in,
                              void* d_out, int out_size, void* d_ws, size_t ws_size,
                              hipStream_t stream) {
    (void)in_sizes; (void)n_in; (void)out_size; (void)ws_size;
    const float* vector = (const float*)d_in[0];
    const float* matrix = (const float*)d_in[1];
    const int*   lens   = (const int*)d_in[2];
    const float* W_vec  = (const float*)d_in[3];
    const float* W_mat  = (const float*)d_in[4];
    const float* w_attn = (const float*)d_in[5];

    float* out = (float*)d_out;
    float* rep = out;                 // [B, DM] first in return order
    float* att = out + B_ * DM_;      // [B, S]

    float* t1     = (float*)d_ws;     // [B, DA]
    float* logits = t1 + B_ * DA_;    // [B, S]
    float* maxv   = logits + B_ * S_; // [1]

    const size_t smemBytes = (size_t)(2 * ABUF_FLOATS + 32) * sizeof(float);

    k_t1    <<<dim3(DA_ / 256, B_), 256, 0,         stream>>>(vector, W_vec, t1);
    k_logits<<<(B_ * S_) / 32,      256, smemBytes, stream>>>(matrix, W_mat, t1, w_attn, logits);
    k_max   <<<1,                  1024, 0,         stream>>>(logits, maxv);
    k_soft  <<<B_,                  512, 0,         stream>>>(logits, maxv, lens, att);
    k_repr  <<<B_ * 4,              256, 0,         stream>>>(matrix, att, rep);
}